// STBlock_42374147342926
// MI455X (gfx1250) — compile-verified
//
#include <hip/hip_runtime.h>
#include <math.h>

// Problem dims (fixed by reference setup)
#define BB 8
#define NN 2048
#define CC 64
#define TT 12
#define KK 3
#define FF 64
#define LN_EPS 1e-5f

typedef float v2f __attribute__((ext_vector_type(2)));
typedef float v8f __attribute__((ext_vector_type(8)));

// fp32 WMMA: D(16x16) = A(16x4) * B(4x16) + C.  8-arg VOP3P form.
__device__ __forceinline__ v8f wmma4(v2f a, v2f b, v8f c) {
    return __builtin_amdgcn_wmma_f32_16x16x4_f32(
        /*neg_a*/false, a, /*neg_b*/false, b,
        /*c_mod*/(short)0, c, /*reuse_a*/false, /*reuse_b*/false);
}

__device__ __forceinline__ float sigmoidf_(float x) {
    return 1.0f / (1.0f + __expf(-x));
}

// ---------------- temporal attention (tiny) ----------------

// tmp[b,t,c] = sum_n x[b,n,c,t] * u1[n]       grid = B*T*C blocks x 256
__global__ void k_tmp_tc(const float* __restrict__ x, const float* __restrict__ u1,
                         float* __restrict__ tmp) {
    int bid = blockIdx.x;
    int b = bid / (TT * CC); int r = bid % (TT * CC);
    int t = r / CC, c = r % CC;
    float acc = 0.f;
    for (int n = threadIdx.x; n < NN; n += 256)
        acc += x[(((size_t)b * NN + n) * CC + c) * TT + t] * u1[n];
    __shared__ float red[256];
    red[threadIdx.x] = acc; __syncthreads();
    for (int s = 128; s > 0; s >>= 1) {
        if (threadIdx.x < s) red[threadIdx.x] += red[threadIdx.x + s];
        __syncthreads();
    }
    if (threadIdx.x == 0) tmp[(b * TT + t) * CC + c] = red[0];
}

// lhs_t[b,t,n] = sum_c tmp[b,t,c]*u2[c,n]     B*T*N threads
__global__ void k_lhs_t(const float* __restrict__ tmp, const float* __restrict__ u2,
                        float* __restrict__ lhst) {
    int idx = blockIdx.x * 256 + threadIdx.x;       // (b*T+t)*N + n
    int bt = idx / NN, n = idx % NN;
    const float* tp = tmp + bt * CC;
    float acc = 0.f;
    for (int c = 0; c < CC; ++c) acc += tp[c] * u2[c * NN + n];
    lhst[idx] = acc;
}

// rhs_t[b,n,t] = sum_c u3[c]*x[b,n,c,t]       B*N*T threads
__global__ void k_rhs_t(const float* __restrict__ x, const float* __restrict__ u3,
                        float* __restrict__ rhst) {
    int idx = blockIdx.x * 256 + threadIdx.x;       // (b*N+n)*T + t
    int bn = idx / TT, t = idx % TT;
    const float* xp = x + (size_t)bn * CC * TT + t;
    float acc = 0.f;
    for (int c = 0; c < CC; ++c) acc += u3[c] * xp[(size_t)c * TT];
    rhst[idx] = acc;
}

// prod_t[b,t,s] = sum_n lhs_t[b,t,n]*rhs_t[b,n,s]   grid = B*T*T blocks x 256
__global__ void k_prod_t(const float* __restrict__ lhst, const float* __restrict__ rhst,
                         float* __restrict__ prodt) {
    int bid = blockIdx.x;
    int b = bid / (TT * TT); int r = bid % (TT * TT);
    int t = r / TT, s2 = r % TT;
    float acc = 0.f;
    for (int n = threadIdx.x; n < NN; n += 256)
        acc += lhst[(b * TT + t) * NN + n] * rhst[((size_t)b * NN + n) * TT + s2];
    __shared__ float red[256];
    red[threadIdx.x] = acc; __syncthreads();
    for (int s = 128; s > 0; s >>= 1) {
        if (threadIdx.x < s) red[threadIdx.x] += red[threadIdx.x + s];
        __syncthreads();
    }
    if (threadIdx.x == 0) prodt[bid] = red[0];
}

// t_att[b,t,u] = softmax_u( sum_s ve[t,s]*sigmoid(prod_t[b,s,u]+be[s,u]) )  grid=B x 160
__global__ void k_tatt(const float* __restrict__ prodt, const float* __restrict__ ve,
                       const float* __restrict__ be, float* __restrict__ tatt) {
    __shared__ float et[TT * TT];
    int b = blockIdx.x, tid = threadIdx.x;
    if (tid < TT * TT) {
        int t = tid / TT, u = tid % TT;
        float e = 0.f;
        for (int s2 = 0; s2 < TT; ++s2) {
            float pv = prodt[(b * TT + s2) * TT + u] + be[s2 * TT + u];
            e += ve[t * TT + s2] * sigmoidf_(pv);
        }
        et[tid] = e;
    }
    __syncthreads();
    if (tid < TT) {
        float m = -1e30f;
        for (int u = 0; u < TT; ++u) m = fmaxf(m, et[tid * TT + u]);
        float ex[TT]; float sum = 0.f;
        for (int u = 0; u < TT; ++u) { ex[u] = __expf(et[tid * TT + u] - m); sum += ex[u]; }
        float inv = 1.f / sum;
        for (int u = 0; u < TT; ++u) tatt[(b * TT + tid) * TT + u] = ex[u] * inv;
    }
}

// x_tatt[b,n,c,u] = sum_t x[b,n,c,t]*t_att[b,t,u]; also q[b,n,c]=sum_u x_tatt*w1[u]
__global__ void k_xtatt(const float* __restrict__ x, const float* __restrict__ tatt,
                        const float* __restrict__ w1, float* __restrict__ xt,
                        float* __restrict__ q) {
    int idx = blockIdx.x * 256 + threadIdx.x;     // b*N*C + n*C + c
    int b = idx / (NN * CC);
    size_t base = (size_t)idx * TT;
    float xr[TT];
    for (int t = 0; t < TT; ++t) xr[t] = x[base + t];
    const float* ta = tatt + b * TT * TT;
    float qa = 0.f;
    for (int u = 0; u < TT; ++u) {
        float o = 0.f;
        for (int t = 0; t < TT; ++t) o += xr[t] * ta[t * TT + u];
        xt[base + u] = o;
        qa += o * w1[u];
    }
    q[idx] = qa;
}

// ---------------- spatial attention ----------------

// lhs_s[b,n,t]=sum_c q[b,n,c]*w2[c,t] ; rhs_s[b,n,t]=sum_c w3[c]*x_tatt[b,n,c,t]
__global__ void k_lrs(const float* __restrict__ q, const float* __restrict__ xt,
                      const float* __restrict__ w2, const float* __restrict__ w3,
                      float* __restrict__ lhs, float* __restrict__ rhs) {
    int idx = blockIdx.x * 256 + threadIdx.x;     // (b*N+n)*T + t
    int bn = idx / TT, t = idx % TT;
    const float* qp = q + (size_t)bn * CC;
    const float* xp = xt + (size_t)bn * CC * TT + t;
    float l = 0.f, r = 0.f;
    for (int c = 0; c < CC; ++c) {
        l += qp[c] * w2[c * TT + t];
        r += w3[c] * xp[(size_t)c * TT];
    }
    lhs[idx] = l; rhs[idx] = r;
}

// sig[b,n,m] = sigmoid( sum_t lhs[b,n,t]*rhs[b,m,t] + bs[n,m] )   grid(128,128,8) x 256
__global__ void __launch_bounds__(256) k_sig(const float* __restrict__ lhs,
                                             const float* __restrict__ rhs,
                                             const float* __restrict__ bs,
                                             float* __restrict__ sig) {
    __shared__ float lt[16][TT], rt[16][TT];
    int b = blockIdx.z, n0 = blockIdx.y * 16, m0 = blockIdx.x * 16;
    int tid = threadIdx.x;
    if (tid < 192) {
        int rr = tid / TT, t2 = tid % TT;
        lt[rr][t2] = lhs[((size_t)b * NN + n0 + rr) * TT + t2];
        rt[rr][t2] = rhs[((size_t)b * NN + m0 + rr) * TT + t2];
    }
    __syncthreads();
    int ln = tid >> 4, lm = tid & 15;
    float acc = bs[(size_t)(n0 + ln) * NN + m0 + lm];
    for (int t = 0; t < TT; ++t) acc += lt[ln][t] * rt[lm][t];
    sig[((size_t)b * NN + n0 + ln) * NN + m0 + lm] = sigmoidf_(acc);
}

// s[b,n,j] = sum_m vs[n,m]*sig[b,m,j]   -- fp32 WMMA GEMM, wave = 16x64 strip.
// Block's 8 waves take CONSECUTIVE n-tiles with the SAME jstrip -> identical B
// loads hit in the WGP$. Barrier every 128 k-steps keeps waves in a 32KB window.
// waves = B * (N/64) * (N/16) = 8*32*128 = 32768 -> 4096 blocks of 8 waves
__global__ void __launch_bounds__(256) k_s_gemm(const float* __restrict__ vs,
                                                const float* __restrict__ sig,
                                                float* __restrict__ s) {
    int wid  = blockIdx.x * 8 + (threadIdx.x >> 5);   // ((b*32+jstrip)*128 + ntile)
    int lane = threadIdx.x & 31;
    int ntile  = wid & 127;
    int rdec   = wid >> 7;
    int jstrip = rdec & 31;
    int b      = rdec >> 5;
    int n0 = ntile * 16, j0 = jstrip * 64;
    bool hi = lane >= 16; int lr = lane & 15;
    int aoff = hi ? 2 : 0;

    v8f acc[4] = {};
    const float* arow  = vs + (size_t)(n0 + lr) * NN + aoff;
    const float* bbase = sig + (size_t)b * NN * NN + j0 + lr;
    for (int mo = 0; mo < NN; mo += 128) {
        for (int m0 = mo; m0 < mo + 128; m0 += 4) {
            float2 av = *(const float2*)(arow + m0);
            v2f a; a.x = av.x; a.y = av.y;
            const float* bp = bbase + (size_t)(m0 + aoff) * NN;
            #pragma unroll
            for (int ct = 0; ct < 4; ++ct) {
                v2f bfrag;
                bfrag.x = bp[ct * 16];
                bfrag.y = bp[ct * 16 + NN];
                acc[ct] = wmma4(a, bfrag, acc[ct]);
            }
        }
        __syncthreads();   // keep the 8 waves' shared-B windows converged
    }
    #pragma unroll
    for (int ct = 0; ct < 4; ++ct)
        #pragma unroll
        for (int g = 0; g < 8; ++g) {
            int n = n0 + g + (hi ? 8 : 0);
            s[((size_t)b * NN + n) * NN + j0 + ct * 16 + lr] = acc[ct][g];
        }
}

// row softmax in place, row = (b*N+n), grid = B*N blocks x 256
__global__ void k_softmax(float* __restrict__ s) {
    float* p = s + (size_t)blockIdx.x * NN;
    __shared__ float red[256];
    int tid = threadIdx.x;
    float m = -1e30f;
    for (int j = tid; j < NN; j += 256) m = fmaxf(m, p[j]);
    red[tid] = m; __syncthreads();
    for (int st = 128; st > 0; st >>= 1) {
        if (tid < st) red[tid] = fmaxf(red[tid], red[tid + st]);
        __syncthreads();
    }
    m = red[0]; __syncthreads();
    float sum = 0.f;
    for (int j = tid; j < NN; j += 256) { float e = __expf(p[j] - m); p[j] = e; sum += e; }
    red[tid] = sum; __syncthreads();
    for (int st = 128; st > 0; st >>= 1) {
        if (tid < st) red[tid] += red[tid + st];
        __syncthreads();
    }
    float inv = 1.f / red[0];
    for (int j = tid; j < NN; j += 256) p[j] *= inv;
}

// ---------------- fused Chebyshev graph conv ----------------
// Wave strip = (b, 16-row n-tile, one t). Columns t-major: strip covers all 64 c.
// Single m-loop holds 3 accumulator sets (one per Chebyshev order), so the
// x_tatt B fragments are loaded ONCE (3x traffic cut). Block's 8 waves share
// (b,t) -> identical B loads hit in WGP$. Then per k: C/D->A via wave-private
// LDS slot, and a small WMMA chain against theta_k accumulates the output.
// waves = B*T*(N/16) = 12288 -> 1536 blocks of 8 waves
__global__ void __launch_bounds__(256) k_gcn(const float* __restrict__ cheb,
                                             const float* __restrict__ satt,
                                             const float* __restrict__ xt,
                                             const float* __restrict__ theta,
                                             float* __restrict__ xgcn) {
    __shared__ float plds[8][16 * 64];
    int wslot = threadIdx.x >> 5;
    int lane  = threadIdx.x & 31;
    int wid = blockIdx.x * 8 + wslot;        // ((b*T + t)*128 + ntile)
    int ntile = wid & 127;
    int bt = wid >> 7;
    int t = bt % TT;
    int b = bt / TT;
    int n0 = ntile * 16;
    bool hi = lane >= 16; int lr = lane & 15;
    int aoff = hi ? 2 : 0;

    v8f pacc[KK][4] = {};
    const float* crow0 = cheb + (size_t)(n0 + lr) * NN + aoff;              // +k*N*N
    const float* srow  = satt + (size_t)b * NN * NN + (size_t)(n0 + lr) * NN + aoff;
    const float* xbase = xt + (size_t)b * NN * CC * TT + t;
    for (int mo = 0; mo < NN; mo += 128) {
        for (int m0 = mo; m0 < mo + 128; m0 += 4) {
            float2 sv = *(const float2*)(srow + m0);
            v2f a[KK];
            #pragma unroll
            for (int k = 0; k < KK; ++k) {
                float2 cv = *(const float2*)(crow0 + (size_t)k * NN * NN + m0);
                a[k].x = cv.x * sv.x; a[k].y = cv.y * sv.y;
            }
            const float* xp0 = xbase + (size_t)(m0 + aoff) * CC * TT;
            const float* xp1 = xbase + (size_t)(m0 + aoff + 1) * CC * TT;
            #pragma unroll
            for (int ct = 0; ct < 4; ++ct) {
                int c = ct * 16 + lr;
                v2f bfrag;
                bfrag.x = xp0[(size_t)c * TT];
                bfrag.y = xp1[(size_t)c * TT];
                #pragma unroll
                for (int k = 0; k < KK; ++k)
                    pacc[k][ct] = wmma4(a[k], bfrag, pacc[k][ct]);
            }
        }
        __syncthreads();   // keep the 8 waves' shared-B windows converged
    }

    v8f oacc[4] = {};
    for (int k = 0; k < KK; ++k) {
        // C/D layout -> row-major 16x64 in this wave's LDS slot
        #pragma unroll
        for (int ct = 0; ct < 4; ++ct)
            #pragma unroll
            for (int g = 0; g < 8; ++g)
                plds[wslot][(g + (hi ? 8 : 0)) * 64 + ct * 16 + lr] = pacc[k][ct][g];
        __syncthreads();
        // out += P_k(16x64) @ theta_k(64x64)
        const float* th = theta + k * CC * FF;
        for (int c0 = 0; c0 < CC; c0 += 4) {
            v2f a2;
            a2.x = plds[wslot][lr * 64 + c0 + aoff];
            a2.y = plds[wslot][lr * 64 + c0 + aoff + 1];
            #pragma unroll
            for (int ct = 0; ct < 4; ++ct) {
                int f = ct * 16 + lr;
                v2f bfrag;
                bfrag.x = th[(c0 + aoff) * FF + f];
                bfrag.y = th[(c0 + aoff + 1) * FF + f];
                oacc[ct] = wmma4(a2, bfrag, oacc[ct]);
            }
        }
        __syncthreads();
    }
    #pragma unroll
    for (int ct = 0; ct < 4; ++ct)
        #pragma unroll
        for (int g = 0; g < 8; ++g) {
            int n = n0 + g + (hi ? 8 : 0);
            int f = ct * 16 + lr;
            xgcn[(((size_t)b * NN + n) * FF + f) * TT + t] = fmaxf(oacc[ct][g], 0.f);
        }
}

// ---------------- time conv + residual + relu + layernorm ----------------
// block per (b,n), 768 threads = (f,t)
__global__ void __launch_bounds__(768) k_final(const float* __restrict__ xgcn,
                                               const float* __restrict__ x,
                                               const float* __restrict__ tw,
                                               const float* __restrict__ tb,
                                               const float* __restrict__ rw,
                                               const float* __restrict__ rb,
                                               const float* __restrict__ lng,
                                               const float* __restrict__ lnb,
                                               float* __restrict__ out) {
    __shared__ float xg[CC * TT], xl[CC * TT], vbuf[TT * FF], smu[TT], srs[TT];
    size_t base = (size_t)blockIdx.x * (CC * TT);
    int tid = threadIdx.x;
    xg[tid] = xgcn[base + tid];
    xl[tid] = x[base + tid];
    __syncthreads();
    int f = tid / TT, t = tid % TT;
    float acc = tb[f] + rb[f];
    for (int c = 0; c < CC; ++c) {
        const float* w = tw + (f * CC + c) * 3;
        float x0 = (t >= 1)     ? xg[c * TT + t - 1] : 0.f;
        float x1 =                xg[c * TT + t];
        float x2 = (t + 1 < TT) ? xg[c * TT + t + 1] : 0.f;
        acc += w[0] * x0 + w[1] * x1 + w[2] * x2;
        acc += rw[f * CC + c] * xl[c * TT + t];
    }
    float v = fmaxf(acc, 0.f);
    vbuf[t * FF + f] = v;
    __syncthreads();
    if (tid < TT) {
        float mu = 0.f;
        for (int ff = 0; ff < FF; ++ff) mu += vbuf[tid * FF + ff];
        mu *= (1.f / FF);
        float var = 0.f;
        for (int ff = 0; ff < FF; ++ff) { float d = vbuf[tid * FF + ff] - mu; var += d * d; }
        var *= (1.f / FF);
        smu[tid] = mu; srs[tid] = rsqrtf(var + LN_EPS);
    }
    __syncthreads();
    out[base + tid] = (v - smu[t]) * srs[t] * lng[f] + lnb[f];
}

// ---------------- host ----------------
extern "C" void kernel_launch(void* const* d_in, const int* in_sizes, int n_in,
                              void* d_out, int out_size, void* d_ws, size_t ws_size,
                              hipStream_t stream) {
    (void)in_sizes; (void)n_in; (void)out_size; (void)ws_size;
    const float* x    = (const float*)d_in[0];
    const float* cheb = (const float*)d_in[1];
    const float* th   = (const float*)d_in[2];
    const float* u1   = (const float*)d_in[3];
    const float* u2   = (const float*)d_in[4];
    const float* u3   = (const float*)d_in[5];
    const float* be   = (const float*)d_in[6];
    const float* ve   = (const float*)d_in[7];
    const float* w1   = (const float*)d_in[8];
    const float* w2   = (const float*)d_in[9];
    const float* w3   = (const float*)d_in[10];
    const float* bs   = (const float*)d_in[11];
    const float* vs   = (const float*)d_in[12];
    const float* tw   = (const float*)d_in[13];
    const float* tb   = (const float*)d_in[14];
    const float* rw   = (const float*)d_in[15];
    const float* rb   = (const float*)d_in[16];
    const float* lng  = (const float*)d_in[17];
    const float* lnb  = (const float*)d_in[18];

    float* ws = (float*)d_ws;
    size_t off = 0;
    auto take = [&](size_t n) { float* p = ws + off; off += n; return p; };
    float* tatt  = take((size_t)BB * TT * TT);
    float* tmp   = take((size_t)BB * TT * CC);
    float* lhst  = take((size_t)BB * TT * NN);
    float* rhst  = take((size_t)BB * NN * TT);
    float* prodt = take((size_t)BB * TT * TT);
    float* xt    = take((size_t)BB * NN * CC * TT);
    float* q     = take((size_t)BB * NN * CC);
    float* lhss  = take((size_t)BB * NN * TT);
    float* rhss  = take((size_t)BB * NN * TT);
    float* sig   = take((size_t)BB * NN * NN);
    float* satt  = take((size_t)BB * NN * NN);
    float* xg    = take((size_t)BB * NN * FF * TT);

    k_tmp_tc <<<BB * TT * CC, 256, 0, stream>>>(x, u1, tmp);
    k_lhs_t  <<<(BB * TT * NN) / 256, 256, 0, stream>>>(tmp, u2, lhst);
    k_rhs_t  <<<(BB * NN * TT) / 256, 256, 0, stream>>>(x, u3, rhst);
    k_prod_t <<<BB * TT * TT, 256, 0, stream>>>(lhst, rhst, prodt);
    k_tatt   <<<BB, 160, 0, stream>>>(prodt, ve, be, tatt);
    k_xtatt  <<<(BB * NN * CC) / 256, 256, 0, stream>>>(x, tatt, w1, xt, q);
    k_lrs    <<<(BB * NN * TT) / 256, 256, 0, stream>>>(q, xt, w2, w3, lhss, rhss);
    k_sig    <<<dim3(NN / 16, NN / 16, BB), 256, 0, stream>>>(lhss, rhss, bs, sig);
    k_s_gemm <<<(BB * (NN / 16) * (NN / 64)) / 8, 256, 0, stream>>>(vs, sig, satt);
    k_softmax<<<BB * NN, 256, 0, stream>>>(satt);
    k_gcn    <<<(BB * (NN / 16) * TT) / 8, 256, 0, stream>>>(cheb, satt, xt, th, xg);
    k_final  <<<BB * NN, 768, 0, stream>>>(xg, x, tw, tb, rw, rb, lng, lnb, (float*)d_out);
}